// TorchGrouper_56719338111369
// MI455X (gfx1250) — compile-verified
//
#include <hip/hip_runtime.h>
#include <hip/hip_bf16.h>
#include <stdint.h>

#ifndef __has_builtin
#define __has_builtin(x) 0
#endif

#if __has_builtin(__builtin_amdgcn_global_load_async_to_lds_b128)
#define ASYNC_LDS 1
#else
#define ASYNC_LDS 0
#endif

// Problem constants (fixed by setup_inputs in the reference)
constexpr int ZD = 40, YD = 400, XD = 400;   // voxel map dims
constexpr int CF = 64;                       // feature channels
constexpr int OF = 64;                       // hollow-cube offsets: p=4 -> 4^3
constexpr int TSTRIDE = 68;                  // padded LDS row stride (floats), keeps 16B alignment

// r = concat(arange(-3,-1), arange(1,3)) = {-3,-2,1,2}
__device__ __forceinline__ int rval(int i) { return (i < 2) ? (i - 3) : (i - 1); }

#if ASYNC_LDS
typedef int v4i __attribute__((ext_vector_type(4)));
typedef __attribute__((address_space(1))) v4i* gv4_t;   // global int4*
typedef __attribute__((address_space(3))) v4i* lv4_t;   // LDS int4*
#endif

__global__ __launch_bounds__(256) void grouper_kernel(
    const int*   __restrict__ voxel_maps,      // [N,Z,Y,X] int32, -1 == empty
    const float* __restrict__ grid_positions,  // [G,4] (b,z,y,x)
    const float* __restrict__ features,        // [M,C]
    float*       __restrict__ sf_out,          // [C,G,O]
    float*       __restrict__ gpf_out,         // [3,G,O]
    float*       __restrict__ mask_out,        // [G]
    int G)
{
    __shared__ float tile[OF * TSTRIDE];   // 64 x 68 floats = 17.4 KB
    __shared__ int   s_idx[OF];
    __shared__ int   s_any;

    const int g   = blockIdx.x;
    const int tid = threadIdx.x;
    const size_t GO = (size_t)G * OF;

    if (tid == 0) s_any = 0;
    __syncthreads();

    // Uniform per-block grid position (scalarizes to s_load)
    const float pb = grid_positions[(size_t)g * 4 + 0];
    const float pz = grid_positions[(size_t)g * 4 + 1];
    const float py = grid_positions[(size_t)g * 4 + 2];
    const float px = grid_positions[(size_t)g * 4 + 3];

    // ---- Phase 1: indices, gpf rows, mask flag (lanes 0..63, one offset each) ----
    if (tid < OF) {
        const int o  = tid;
        const int zo = rval(o & 3);          // offsets col1 (added to z)
        const int yo = rval((o >> 2) & 3);   // offsets col2 (added to y)
        const int xo = rval((o >> 4) & 3);   // offsets col3 (added to x)

        const float fz = pz + (float)zo;
        const float fy = py + (float)yo;
        const float fx = px + (float)xo;

        const int bi = (int)pb;                                  // trunc toward zero, unclamped (matches ref)
        int zi = (int)fz; zi = min(max(zi, 0), ZD - 1);
        int yi = (int)fy; yi = min(max(yi, 0), YD - 1);
        int xi = (int)fx; xi = min(max(xi, 0), XD - 1);

        const int idx = voxel_maps[(((size_t)bi * ZD + zi) * YD + yi) * XD + xi];
        s_idx[o] = idx;
        if (idx >= 0) {
            atomicOr(&s_any, 1);
            // Warm L2 for the data-dependent feature row (emits global_prefetch_b8)
            __builtin_prefetch(features + (size_t)idx * CF, 0, 3);
        }

        // gpf: frac part + faithful quirk (row z += 0, row y += z-offset, row x += y-offset)
        const size_t base = (size_t)g * OF + o;
        gpf_out[base]          =  fz - truncf(fz);
        gpf_out[GO + base]     = (fy - truncf(fy)) + (float)zo;
        gpf_out[2 * GO + base] = (fx - truncf(fx)) + (float)yo;
    }
    __syncthreads();

    if (tid == 0) mask_out[g] = (s_any == 0) ? 1.0f : 0.0f;

    // ---- Phase 2: gather 64 feature rows into LDS tile (4 threads per row) ----
    {
        const int o   = tid >> 2;
        const int seg = (tid & 3) * 16;      // 16 floats (64B) per thread
        const int idx = s_idx[o];
        float* dst = &tile[o * TSTRIDE + seg];
        if (idx >= 0) {
            const float* src = features + (size_t)idx * CF + seg;
#if ASYNC_LDS
            __builtin_amdgcn_global_load_async_to_lds_b128((gv4_t)(src +  0), (lv4_t)(dst +  0), 0, 0);
            __builtin_amdgcn_global_load_async_to_lds_b128((gv4_t)(src +  4), (lv4_t)(dst +  4), 0, 0);
            __builtin_amdgcn_global_load_async_to_lds_b128((gv4_t)(src +  8), (lv4_t)(dst +  8), 0, 0);
            __builtin_amdgcn_global_load_async_to_lds_b128((gv4_t)(src + 12), (lv4_t)(dst + 12), 0, 0);
#else
            float4 v0 = ((const float4*)src)[0];
            float4 v1 = ((const float4*)src)[1];
            float4 v2 = ((const float4*)src)[2];
            float4 v3 = ((const float4*)src)[3];
            ((float4*)dst)[0] = v0; ((float4*)dst)[1] = v1;
            ((float4*)dst)[2] = v2; ((float4*)dst)[3] = v3;
#endif
        } else {
            // feats_pad row 0 == zeros
            const float4 z4 = make_float4(0.f, 0.f, 0.f, 0.f);
            ((float4*)dst)[0] = z4; ((float4*)dst)[1] = z4;
            ((float4*)dst)[2] = z4; ((float4*)dst)[3] = z4;
        }
    }
#if ASYNC_LDS
#if __has_builtin(__builtin_amdgcn_s_wait_asynccnt)
    __builtin_amdgcn_s_wait_asynccnt(0);
#else
    asm volatile("s_wait_asynccnt 0" ::: "memory");
#endif
#endif
    __syncthreads();

    // ---- Phase 3: transposed, fully-coalesced write-out: out[c*G*O + g*O + o] ----
    {
        const int o  = tid & 63;
        const int cs = tid >> 6;             // 0..3
        float* outp = sf_out + (size_t)g * OF + o;
        #pragma unroll
        for (int cb = 0; cb < CF; cb += 4) {
            const int c = cb + cs;
            outp[(size_t)c * GO] = tile[o * TSTRIDE + c];
        }
    }
}

extern "C" void kernel_launch(void* const* d_in, const int* in_sizes, int n_in,
                              void* d_out, int out_size, void* d_ws, size_t ws_size,
                              hipStream_t stream) {
    const int*   voxel_maps     = (const int*)d_in[0];
    const float* grid_positions = (const float*)d_in[1];
    const float* features       = (const float*)d_in[2];

    const int G = in_sizes[1] / 4;           // grid_positions is [G,4]

    float* sf_out   = (float*)d_out;                      // C*G*O floats
    float* gpf_out  = sf_out  + (size_t)CF * G * OF;      // 3*G*O floats
    float* mask_out = gpf_out + (size_t)3  * G * OF;      // G floats (bool -> 0/1)

    grouper_kernel<<<dim3(G), dim3(256), 0, stream>>>(
        voxel_maps, grid_positions, features, sf_out, gpf_out, mask_out, G);
}